// GNN_branch_with_GAT_60954175865310
// MI455X (gfx1250) — compile-verified
//
#include <hip/hip_runtime.h>
#include <math.h>

typedef __attribute__((ext_vector_type(2))) float v2f;
typedef __attribute__((ext_vector_type(8))) float v8f;

#define NNODES 20000
#define NEDGES 160000
#define NGRAPH 64
#define ETOT   (NEDGES + NNODES)   // edges + appended self loops
#define EPSN   1e-5f

// ---------------------------------------------------------------- utilities

__device__ __forceinline__ void atomic_max_float(float* addr, float val) {
  // signed-int max for val>=0, unsigned-int min for val<0; init must be -inf
  if (val >= 0.0f) {
    atomicMax((int*)addr, __float_as_int(val));
  } else {
    atomicMin((unsigned int*)addr, (unsigned int)__float_as_int(val));
  }
}

__global__ __launch_bounds__(256) void k_fill(float* __restrict__ p, float v, int n) {
  int i = blockIdx.x * 256 + threadIdx.x;
  if (i < n) p[i] = v;
}

__global__ __launch_bounds__(256) void k_counts(const int* __restrict__ batch,
                                                float* __restrict__ counts, int n) {
  int i = blockIdx.x * 256 + threadIdx.x;
  if (i < n) atomicAdd(&counts[batch[i]], 1.0f);
}

// zero-padded copy: src [K,N] -> dst [Kpad,Npad]
__global__ __launch_bounds__(256) void k_pad_copy(const float* __restrict__ src,
                                                  float* __restrict__ dst,
                                                  int K, int N, int Npad, int total) {
  int i = blockIdx.x * 256 + threadIdx.x;
  if (i >= total) return;
  int r = i / Npad, c = i - r * Npad;
  dst[i] = (r < K && c < N) ? src[r * N + c] : 0.0f;
}

// ------------------------------------------------- fp32 WMMA GEMM  C = A @ B
// A: [M,K] row-major (K multiple of 4, M multiple of 16), B: [K,Npad] zero-padded
// row-major (Npad multiple of 64), C: [M,N] (stores guarded on col<N only).
// One wave computes a 16x64 strip: 4 independent 16x16 WMMA accumulators.
// V_WMMA_F32_16X16X4_F32 lane layout (ISA 7.12.2):
//   A 16x4 : lanes 0-15 rows M=0..15 hold K={k0,k0+1}; lanes 16-31 K={k0+2,k0+3}
//   B 4x16 : lanes 0-15 cols N=0..15 hold K={k0,k0+1}; lanes 16-31 K={k0+2,k0+3}
//   C/D    : VGPR i -> row (i + 8*half), col = lane&15

__global__ __launch_bounds__(256) void k_gemm_f32_wmma(const float* __restrict__ A,
                                                       const float* __restrict__ B,
                                                       float* __restrict__ C,
                                                       int M, int K, int Npad, int N) {
  const int lane   = threadIdx.x & 31;
  const int wave   = threadIdx.x >> 5;
  const int strips = Npad >> 6;                 // 16x64 strips per row band
  const int tile   = blockIdx.x * 8 + wave;
  if (tile >= (M >> 4) * strips) return;        // wave-uniform exit
  const int tm   = tile / strips;
  const int ts   = tile - tm * strips;
  const int half = lane >> 4;                   // 0 or 1
  const int l16  = lane & 15;
  const float* __restrict__ Arow = A + (size_t)(tm * 16 + l16) * K;
  const int bcol = ts * 64 + l16;

  v8f acc0 = {}, acc1 = {}, acc2 = {}, acc3 = {};
  for (int k0 = 0; k0 < K; k0 += 4) {
    const int ka = k0 + half * 2;
    // speculative prefetch 4 steps ahead (invalid translations silently dropped)
    __builtin_prefetch(&Arow[k0 + 16], 0, 3);
    __builtin_prefetch(&B[(size_t)(k0 + 16) * Npad + bcol], 0, 3);
    const v2f a = *(const v2f*)(Arow + ka);     // 8B-aligned: K,ka even
    const float* __restrict__ B0 = B + (size_t)ka * Npad + bcol;
    const float* __restrict__ B1 = B0 + Npad;
    v2f b;
    b.x = B0[0];  b.y = B1[0];
    acc0 = __builtin_amdgcn_wmma_f32_16x16x4_f32(false, a, false, b, (short)0, acc0, false, false);
    b.x = B0[16]; b.y = B1[16];
    acc1 = __builtin_amdgcn_wmma_f32_16x16x4_f32(false, a, false, b, (short)0, acc1, false, false);
    b.x = B0[32]; b.y = B1[32];
    acc2 = __builtin_amdgcn_wmma_f32_16x16x4_f32(false, a, false, b, (short)0, acc2, false, false);
    b.x = B0[48]; b.y = B1[48];
    acc3 = __builtin_amdgcn_wmma_f32_16x16x4_f32(false, a, false, b, (short)0, acc3, false, false);
  }

  const int r0 = tm * 16 + half * 8;            // M multiple of 16: no row guard
  v8f accs[4] = {acc0, acc1, acc2, acc3};
#pragma unroll
  for (int j = 0; j < 4; ++j) {
    const int col = ts * 64 + j * 16 + l16;
    if (col < N) {
#pragma unroll
      for (int i = 0; i < 8; ++i) C[(size_t)(r0 + i) * N + col] = accs[j][i];
    }
  }
}

// ------------------------------------------------- attention dot products
// one wave32 per node: s_src[i] = hp[i].asrc ; s_dst[i] = hp[i].adst
__global__ __launch_bounds__(256) void k_dots(const float* __restrict__ hp,
                                              const float* __restrict__ asrc,
                                              const float* __restrict__ adst,
                                              float* __restrict__ ssrc,
                                              float* __restrict__ sdst,
                                              int n, int D) {
  const int node = (blockIdx.x * 256 + threadIdx.x) >> 5;
  const int lane = threadIdx.x & 31;
  if (node >= n) return;
  const float* __restrict__ row = hp + (size_t)node * D;
  float a = 0.0f, b = 0.0f;
  for (int c = lane; c < D; c += 32) {
    const float v = row[c];
    a += v * asrc[c];
    b += v * adst[c];
  }
#pragma unroll
  for (int off = 16; off > 0; off >>= 1) {
    a += __shfl_xor(a, off, 32);
    b += __shfl_xor(b, off, 32);
  }
  if (lane == 0) { ssrc[node] = a; sdst[node] = b; }
}

// ------------------------------------------------- edge softmax passes
__device__ __forceinline__ void edge_sd(const int* __restrict__ ei, int e,
                                        int* s, int* d) {
  if (e < NEDGES) { *s = ei[e]; *d = ei[NEDGES + e]; }
  else            { *s = *d = e - NEDGES; }          // appended self loop
}

__global__ __launch_bounds__(256) void k_edge_max(const int* __restrict__ ei,
                                                  const float* __restrict__ ssrc,
                                                  const float* __restrict__ sdst,
                                                  float* __restrict__ m) {
  const int e = blockIdx.x * 256 + threadIdx.x;
  if (e >= ETOT) return;
  int s, d; edge_sd(ei, e, &s, &d);
  float v = ssrc[s] + sdst[d];
  v = (v >= 0.0f) ? v : 0.2f * v;                    // attention LeakyReLU
  atomic_max_float(&m[d], v);
}

__global__ __launch_bounds__(256) void k_fix_m(float* __restrict__ m, int n) {
  const int i = blockIdx.x * 256 + threadIdx.x;
  if (i < n && !__builtin_isfinite(m[i])) m[i] = 0.0f;
}

__global__ __launch_bounds__(256) void k_edge_expsum(const int* __restrict__ ei,
                                                     const float* __restrict__ ssrc,
                                                     const float* __restrict__ sdst,
                                                     const float* __restrict__ m,
                                                     float* __restrict__ ex,
                                                     float* __restrict__ denom) {
  const int e = blockIdx.x * 256 + threadIdx.x;
  if (e >= ETOT) return;
  int s, d; edge_sd(ei, e, &s, &d);
  float v = ssrc[s] + sdst[d];
  v = (v >= 0.0f) ? v : 0.2f * v;
  const float x = expf(v - m[d]);
  ex[e] = x;
  atomicAdd(&denom[d], x);
}

// out[dst] += alpha * hp[src]   (one block per edge, threads stride channels)
__global__ __launch_bounds__(128) void k_edge_agg(const int* __restrict__ ei,
                                                  const float* __restrict__ ex,
                                                  const float* __restrict__ denom,
                                                  const float* __restrict__ hp,
                                                  float* __restrict__ out, int D) {
  const int e = blockIdx.x;
  int s, d; edge_sd(ei, e, &s, &d);
  const float alpha = ex[e] / (denom[d] + 1e-16f);
  const float* __restrict__ hs = hp  + (size_t)s * D;
  float* __restrict__       od = out + (size_t)d * D;
  for (int c = threadIdx.x; c < D; c += 128) atomicAdd(&od[c], alpha * hs[c]);
}

// ------------------------------------------------- bias + LeakyReLU + GraphNorm
__global__ __launch_bounds__(256) void k_bias_act_sum(float* __restrict__ h,
                                                      const float* __restrict__ bias,
                                                      const int* __restrict__ batch,
                                                      float* __restrict__ sums,
                                                      int n, int D) {
  const int i = blockIdx.x * 256 + threadIdx.x;
  if (i >= n * D) return;
  const int node = i / D, c = i - node * D;
  float v = h[i] + bias[c];
  v = (v >= 0.0f) ? v : 0.01f * v;                   // nn.LeakyReLU
  h[i] = v;
  atomicAdd(&sums[batch[node] * D + c], v);
}

__global__ __launch_bounds__(256) void k_center_sq(float* __restrict__ h,
                                                   const int* __restrict__ batch,
                                                   const float* __restrict__ counts,
                                                   const float* __restrict__ sums,
                                                   const float* __restrict__ ms,
                                                   float* __restrict__ sq,
                                                   int n, int D) {
  const int i = blockIdx.x * 256 + threadIdx.x;
  if (i >= n * D) return;
  const int node = i / D, c = i - node * D;
  const int g = batch[node];
  const float mean = sums[g * D + c] / counts[g];
  const float o = h[i] - ms[c] * mean;
  h[i] = o;
  atomicAdd(&sq[g * D + c], o * o);
}

__global__ __launch_bounds__(256) void k_norm_final(float* __restrict__ h,
                                                    const int* __restrict__ batch,
                                                    const float* __restrict__ counts,
                                                    const float* __restrict__ sq,
                                                    const float* __restrict__ gam,
                                                    const float* __restrict__ bet,
                                                    int n, int D) {
  const int i = blockIdx.x * 256 + threadIdx.x;
  if (i >= n * D) return;
  const int node = i / D, c = i - node * D;
  const int g = batch[node];
  const float var = sq[g * D + c] / counts[g];
  h[i] = h[i] * (1.0f / sqrtf(var + EPSN)) * gam[c] + bet[c];
}

// ------------------------------------------------- readout (mean & max pool)
__global__ __launch_bounds__(256) void k_pool_init(float* __restrict__ out, int D) {
  const int i = blockIdx.x * 256 + threadIdx.x;
  if (i >= NGRAPH * 2 * D) return;
  const int slot = (i / D) & 1;                      // 0 = mean, 1 = max
  out[i] = slot ? -INFINITY : 0.0f;
}

__global__ __launch_bounds__(256) void k_pool(const float* __restrict__ h,
                                              const int* __restrict__ batch,
                                              float* __restrict__ out, int n, int D) {
  const int i = blockIdx.x * 256 + threadIdx.x;
  if (i >= n * D) return;
  const int node = i / D, c = i - node * D;
  const int g = batch[node];
  const float v = h[i];
  atomicAdd(&out[(size_t)g * 2 * D + c], v);
  atomic_max_float(&out[(size_t)g * 2 * D + D + c], v);
}

__global__ __launch_bounds__(256) void k_pool_finish(float* __restrict__ out,
                                                     const float* __restrict__ counts,
                                                     int D) {
  const int i = blockIdx.x * 256 + threadIdx.x;
  if (i >= NGRAPH * D) return;
  const int g = i / D, c = i - g * D;
  out[(size_t)g * 2 * D + c] /= counts[g];
}

// ---------------------------------------------------------------- driver

static inline int cdiv(long a, long b) { return (int)((a + b - 1) / b); }

extern "C" void kernel_launch(void* const* d_in, const int* in_sizes, int n_in,
                              void* d_out, int out_size, void* d_ws, size_t ws_size,
                              hipStream_t stream) {
  (void)in_sizes; (void)n_in; (void)out_size; (void)ws_size;
  const float* x     = (const float*)d_in[0];
  const int*   ei    = (const int*)  d_in[1];
  const int*   batch = (const int*)  d_in[2];
  struct Layer { const float *W, *asrc, *adst, *b, *g, *bt, *ms; };
  Layer L[4];
  for (int l = 0; l < 4; ++l) {
    const int base = 3 + l * 7;
    L[l].W    = (const float*)d_in[base + 0];
    L[l].asrc = (const float*)d_in[base + 1];
    L[l].adst = (const float*)d_in[base + 2];
    L[l].b    = (const float*)d_in[base + 3];
    L[l].g    = (const float*)d_in[base + 4];
    L[l].bt   = (const float*)d_in[base + 5];
    L[l].ms   = (const float*)d_in[base + 6];
  }
  const int dims[5]  = {9, 128, 256, 512, 1534};
  const int kpads[4] = {12, 128, 256, 512};     // K rounded to 4
  const int npads[4] = {128, 256, 512, 1536};   // dout rounded to 64
  const int DMAX = 1534;

  // workspace carve (bump allocator, 256B aligned)
  char* p = (char*)d_ws;
  auto carve = [&](size_t bytes) -> float* {
    float* r = (float*)p;
    p += (bytes + 255) & ~(size_t)255;
    return r;
  };
  float* bufA   = carve((size_t)NNODES * DMAX * 4);  // agg out / layer input
  float* bufB   = carve((size_t)NNODES * DMAX * 4);  // hp = h @ W
  float* Wp     = carve((size_t)512 * 1536 * 4);     // padded weights (per layer)
  float* xpad   = carve((size_t)NNODES * 12 * 4);    // x padded to K=12
  float* ssrc   = carve(NNODES * 4);
  float* sdst   = carve(NNODES * 4);
  float* m      = carve(NNODES * 4);
  float* denom  = carve(NNODES * 4);
  float* ex     = carve(ETOT * 4);
  float* sums   = carve(NGRAPH * DMAX * 4);
  float* sq     = carve(NGRAPH * DMAX * 4);
  float* counts = carve(NGRAPH * 4);

  // per-graph node counts
  k_fill<<<1, 256, 0, stream>>>(counts, 0.0f, NGRAPH);
  k_counts<<<cdiv(NNODES, 256), 256, 0, stream>>>(batch, counts, NNODES);

  // pad x: [20000,9] -> [20000,12]
  k_pad_copy<<<cdiv(NNODES * 12, 256), 256, 0, stream>>>(x, xpad, NNODES, 9, 12,
                                                         NNODES * 12);

  const float* hin = xpad;
  for (int l = 0; l < 4; ++l) {
    const int din = dims[l], dout = dims[l + 1];
    const int Kit = kpads[l], Npad = npads[l];
    const int ND = NNODES * dout;

    // 0) pad W: [din,dout] -> [Kit,Npad] (zero-filled)
    k_pad_copy<<<cdiv(Kit * Npad, 256), 256, 0, stream>>>(L[l].W, Wp, din, dout,
                                                          Npad, Kit * Npad);
    // 1) hp = h @ W   (fp32 WMMA, 16x64 strip per wave, no inner-loop guards)
    const int tiles = (NNODES / 16) * (Npad / 64);
    k_gemm_f32_wmma<<<cdiv(tiles, 8), 256, 0, stream>>>(hin, Wp, bufB,
                                                        NNODES, Kit, Npad, dout);
    // 2) per-node attention logits
    k_dots<<<cdiv((long)NNODES * 32, 256), 256, 0, stream>>>(bufB, L[l].asrc,
                                                             L[l].adst, ssrc, sdst,
                                                             NNODES, dout);
    // 3) segment softmax over dst + scatter-add aggregation
    k_fill<<<cdiv(NNODES, 256), 256, 0, stream>>>(m, -INFINITY, NNODES);
    k_fill<<<cdiv(NNODES, 256), 256, 0, stream>>>(denom, 0.0f, NNODES);
    k_fill<<<cdiv(ND, 256), 256, 0, stream>>>(bufA, 0.0f, ND);
    k_edge_max<<<cdiv(ETOT, 256), 256, 0, stream>>>(ei, ssrc, sdst, m);
    k_fix_m<<<cdiv(NNODES, 256), 256, 0, stream>>>(m, NNODES);
    k_edge_expsum<<<cdiv(ETOT, 256), 256, 0, stream>>>(ei, ssrc, sdst, m, ex, denom);
    k_edge_agg<<<ETOT, 128, 0, stream>>>(ei, ex, denom, bufB, bufA, dout);

    // 4) bias + LeakyReLU(0.01) + GraphNorm (3 passes over bufA in place)
    k_fill<<<cdiv(NGRAPH * dout, 256), 256, 0, stream>>>(sums, 0.0f, NGRAPH * dout);
    k_fill<<<cdiv(NGRAPH * dout, 256), 256, 0, stream>>>(sq, 0.0f, NGRAPH * dout);
    k_bias_act_sum<<<cdiv(ND, 256), 256, 0, stream>>>(bufA, L[l].b, batch, sums,
                                                      NNODES, dout);
    k_center_sq<<<cdiv(ND, 256), 256, 0, stream>>>(bufA, batch, counts, sums,
                                                   L[l].ms, sq, NNODES, dout);
    k_norm_final<<<cdiv(ND, 256), 256, 0, stream>>>(bufA, batch, counts, sq,
                                                    L[l].g, L[l].bt, NNODES, dout);
    hin = bufA;
  }

  // readout: [G, 2, 1534] = concat(mean, max)
  float* out = (float*)d_out;
  const int D = dims[4];
  k_pool_init<<<cdiv(NGRAPH * 2 * D, 256), 256, 0, stream>>>(out, D);
  k_pool<<<cdiv(NNODES * D, 256), 256, 0, stream>>>(bufA, batch, out, NNODES, D);
  k_pool_finish<<<cdiv(NGRAPH * D, 256), 256, 0, stream>>>(out, counts, D);
}